// PatchedGroup_37598143709553
// MI455X (gfx1250) — compile-verified
//
#include <hip/hip_runtime.h>
#include <hip/hip_bf16.h>
#include <float.h>

#define B_ 8
#define N_ 16384
#define C_ 6
#define G_ 512
#define K_ 32          // GROUP_SIZE == wave32 width
#define TILE_C 16      // centers per workgroup (WMMA M)
#define CHUNK 256      // points per chunk (16 waves x 16 points)
#define LDSTRIDE (CHUNK + 2)

typedef __attribute__((ext_vector_type(2))) float v2f;
typedef __attribute__((ext_vector_type(8))) float v8f;

__device__ __forceinline__ float readlane_f(float v, int l) {
    return __int_as_float(__builtin_amdgcn_readlane(__float_as_int(v), l));
}

// ---------------- Kernel 1: farthest point sampling (one WG per batch) ---------------
__global__ void __launch_bounds__(1024)
fps_kernel(const float* __restrict__ xyz, int* __restrict__ fpsIdx,
           float* __restrict__ centerOut)
{
    const int b    = blockIdx.x;
    const int tid  = threadIdx.x;
    const int lane = tid & 31;
    const int wid  = tid >> 5;

    __shared__ float redV[32];
    __shared__ int   redI[32];
    __shared__ int   sIdx[G_];
    __shared__ int   sLast;

    const float* base = xyz + (size_t)b * N_ * C_;

    // cache this thread's 16 points + running min-dist in registers
    float px[16], py[16], pz[16], md[16];
#pragma unroll
    for (int k = 0; k < 16; ++k) {
        const float* q = base + (size_t)(tid + (k << 10)) * C_;
        px[k] = q[0]; py[k] = q[1]; pz[k] = q[2];
        md[k] = FLT_MAX;
    }
    if (tid == 0) sIdx[0] = 0;

    int last = 0;
    for (int it = 1; it < G_; ++it) {
        // broadcast coords of last selected point via L0 (all lanes same addr)
        const float* q = base + (size_t)last * C_;
        float lx = q[0], ly = q[1], lz = q[2];

        float bv = -1.0f; int bi = 0x7fffffff;
#pragma unroll
        for (int k = 0; k < 16; ++k) {
            float dx = px[k] - lx, dy = py[k] - ly, dz = pz[k] - lz;
            float d  = dx * dx + dy * dy + dz * dz;
            md[k] = fminf(md[k], d);
            if (md[k] > bv) { bv = md[k]; bi = tid + (k << 10); }
        }
        // wave32 argmax (ties -> smaller index, matching jnp.argmax)
#pragma unroll
        for (int off = 16; off > 0; off >>= 1) {
            float ov = __shfl_xor(bv, off);
            int   oi = __shfl_xor(bi, off);
            if (ov > bv || (ov == bv && oi < bi)) { bv = ov; bi = oi; }
        }
        if (lane == 0) { redV[wid] = bv; redI[wid] = bi; }
        __syncthreads();
        if (wid == 0) {
            float v = redV[lane]; int i = redI[lane];
#pragma unroll
            for (int off = 16; off > 0; off >>= 1) {
                float ov = __shfl_xor(v, off);
                int   oi = __shfl_xor(i, off);
                if (ov > v || (ov == v && oi < i)) { v = ov; i = oi; }
            }
            if (lane == 0) { sLast = i; sIdx[it] = i; }
        }
        __syncthreads();
        last = sLast;
    }
    __syncthreads();

    if (tid < G_) {
        int i = sIdx[tid];
        fpsIdx[b * G_ + tid] = i;
        const float* q = base + (size_t)i * C_;
        float* co = centerOut + ((size_t)b * G_ + tid) * 3;
        co[0] = q[0]; co[1] = q[1]; co[2] = q[2];
    }
}

// ------- Kernel 2: WMMA metric tiles + wave32 distributed top-32 selection -------
// metric D[m][n] = |p_n|^2 - 2 c_m . p_n  (== d2 minus per-row const |c_m|^2,
// which preserves per-center top-k ordering)
__global__ void __launch_bounds__(512)
knn_kernel(const float* __restrict__ xyz, const float* __restrict__ centerF,
           int* __restrict__ idxOut)
{
    const int b    = blockIdx.x >> 5;     // 32 center-tiles per batch
    const int tile = blockIdx.x & 31;
    const int g0   = tile * TILE_C;
    const int tid  = threadIdx.x;
    const int lane = tid & 31;
    const int w    = tid >> 5;            // 16 waves: wave w owns center g0+w

    __shared__ float d2s[TILE_C][LDSTRIDE];

    const float* base = xyz + (size_t)b * N_ * C_;
    const int m  = lane & 15;
    const int hi = lane >> 4;

    // constant A fragment: 16x4 rows [cx, cy, cz, 1]
    const float* cp = centerF + ((size_t)b * G_ + g0 + m) * 3;
    float cx = cp[0], cy = cp[1], cz = cp[2];
    v2f afrag;
    afrag[0] = hi ? cz   : cx;   // K=0 (lanes 0-15) / K=2 (lanes 16-31)
    afrag[1] = hi ? 1.0f : cy;   // K=1 / K=3

    // running top-32: one sorted element per lane (ascending across lanes)
    float val = FLT_MAX;
    int   idx = 0;

    // software pipeline: preload chunk 0's point coords
    float x, y, z;
    {
        const float* q0 = base + (size_t)(w * 16 + m) * C_;
        x = q0[0]; y = q0[1]; z = q0[2];
    }

    for (int chunk = 0; chunk < N_; chunk += CHUNK) {
        // unconditional prefetch with wrapped address (no phi / zero-fill):
        // last iteration's prefetch re-reads chunk 0 and is discarded.
        int nxt = (chunk + CHUNK) & (N_ - 1);
        const float* qn = base + (size_t)(nxt + w * 16 + m) * C_;
        float nx = qn[0], ny = qn[1], nz = qn[2];

        // each wave: one 16(center) x 16(point) metric tile via WMMA
        float pp = x * x + y * y + z * z;
        v2f bfrag;
        bfrag[0] = hi ? (-2.0f * z) : (-2.0f * x);  // K=0 / K=2 cols
        bfrag[1] = hi ? pp          : (-2.0f * y);  // K=1 / K=3 cols
        v8f acc = {};
        acc = __builtin_amdgcn_wmma_f32_16x16x4_f32(false, afrag, false, bfrag,
                                                    (short)0, acc, false, false);
#pragma unroll
        for (int r = 0; r < 8; ++r)
            d2s[r + (hi ? 8 : 0)][w * 16 + m] = acc[r];
        __syncthreads();

        // wave w merges its 256 candidates into its sorted top-32.
        // broadcasts use v_readlane (uniform lane selects) -> scalar predicates,
        // only the insertion lane-shift needs ds_bpermute.
        float kth = readlane_f(val, 31);
#pragma unroll
        for (int j = 0; j < CHUNK / 32; ++j) {
            float vnew = d2s[w][j * 32 + lane];
            int   inew = chunk + j * 32 + lane;
            unsigned cand = (unsigned)__ballot(vnew < kth);
            while (cand) {                       // uniform loop, usually empty
                int src = __ffs(cand) - 1;
                cand &= cand - 1;
                float v = readlane_f(vnew, src);
                int   i = __builtin_amdgcn_readlane(inew, src);
                if (v < kth) {                   // scalar predicate
                    unsigned le = (unsigned)__ballot(val <= v);
                    int pos = __popc(le);
                    float upv = __shfl_up(val, 1);
                    int   upi = __shfl_up(idx, 1);
                    if (lane > pos)       { val = upv; idx = upi; }
                    else if (lane == pos) { val = v;   idx = i;   }
                    kth = readlane_f(val, 31);
                }
            }
        }
        __syncthreads();
        x = nx; y = ny; z = nz;
    }
    // lane l holds the l-th nearest neighbor index (ascending d2 == top_k order)
    idxOut[((size_t)b * G_ + g0 + w) * K_ + lane] = idx;
}

// ---------------- Kernel 3: gather neighborhoods, recenter xyz ----------------
__global__ void gather_kernel(const float* __restrict__ xyz,
                              const float* __restrict__ centerF,
                              const int* __restrict__ idxIn,
                              float* __restrict__ neighOut)
{
    int t = blockIdx.x * blockDim.x + threadIdx.x;   // over B*G*K
    if (t >= B_ * G_ * K_) return;
    int g = (t / K_) % G_;
    int b = t / (K_ * G_);
    int i = idxIn[t];
    // rows are 24B => 8-byte aligned: move as float2 (global b64)
    const float2* q2 = (const float2*)(xyz + ((size_t)b * N_ + i) * C_);
    float2 q01 = q2[0], q23 = q2[1], q45 = q2[2];
    const float* c = centerF + ((size_t)b * G_ + g) * 3;
    float2* o2 = (float2*)(neighOut + (size_t)t * C_);
    o2[0] = make_float2(q01.x - c[0], q01.y - c[1]);
    o2[1] = make_float2(q23.x - c[2], q23.y);
    o2[2] = q45;
}

extern "C" void kernel_launch(void* const* d_in, const int* in_sizes, int n_in,
                              void* d_out, int out_size, void* d_ws, size_t ws_size,
                              hipStream_t stream)
{
    const float* xyz = (const float*)d_in[0];
    float* F = (float*)d_out;
    float* neighOut  = F;                                         // B*G*K*C f32
    float* centerOut = F + (size_t)B_ * G_ * K_ * C_;             // B*G*3  f32
    int*   idxOut    = (int*)(centerOut + (size_t)B_ * G_ * 3);   // B*G*K  i32
    int*   fpsIdx    = (int*)d_ws;                                // B*G    i32

    fps_kernel<<<B_, 1024, 0, stream>>>(xyz, fpsIdx, centerOut);
    knn_kernel<<<B_ * 32, 512, 0, stream>>>(xyz, centerOut, idxOut);
    int tot = B_ * G_ * K_;
    gather_kernel<<<(tot + 255) / 256, 256, 0, stream>>>(xyz, centerOut, idxOut, neighOut);
}